// GIN_52089363366042
// MI455X (gfx1250) — compile-verified
//
#include <hip/hip_runtime.h>

// ---------------- problem constants (match reference) ----------------
#define N_NODES   100000      // 20 * 5000
#define E_EDGES   1600000
#define HD        64          // F_IN == H == EMB == 64
#define SLOPE     0.01f
#define BN_EPS    1e-5f
#define ZPAD      66          // LDS row pitch (floats): 66 % 64 = 2 -> conflict-free col reads

typedef __attribute__((ext_vector_type(2))) float v2f;
typedef __attribute__((ext_vector_type(8))) float v8f;
typedef __attribute__((ext_vector_type(4))) unsigned int v4u;
typedef __attribute__((ext_vector_type(8))) int v8i;
typedef __attribute__((ext_vector_type(4))) int v4i;

// =====================================================================
// Tensor Data Mover: 2D tile load (64 cols x up to 64 rows of f32) into LDS.
// D# built per CDNA5 ISA ch.8; this toolchain exposes the 6-arg builtin
// (g0 uint32x4, g1 int32x8, g2 int32x4, g3 int32x4, g4 int32x8, cpol).
//   pad66=true adds the TDM LDS-pad (every 64 DWORDs, +2 DWORDs) so the tile
//   lands with a 66-float row pitch == ZPAD (bank-conflict-free WMMA reads).
//   rows_avail < 64 -> TDM OOB reads return zero == tail-row zero padding.
// =====================================================================
__device__ __forceinline__ void tdm_load_tile(void* lds_dst, const float* gsrc,
                                              unsigned rows_avail, bool pad66)
{
    const unsigned long long ga = (unsigned long long)(uintptr_t)gsrc;
    v4u g0;
    g0.x = 1u;                                      // count=1 (valid descriptor)
    g0.y = (unsigned)(uintptr_t)lds_dst;            // wave-relative LDS byte addr
    g0.z = (unsigned)ga;                            // global_addr[31:0]
    g0.w = (unsigned)((ga >> 32) & 0x01FFFFFFu)     // global_addr[56:32]
         | 0x80000000u;                             // type=2 ("image")
    unsigned w0 = (2u << 16);                       // data_size = 4 bytes
    if (pad66) w0 |= (1u << 20)                     // pad_enable
              | (5u << 22)                          // pad_interval: every 64 DWORDs
              | (1u << 25);                         // pad_amount: +2 DWORDs
    v8i g1;
    g1[0] = (int)w0;                                // wg_mask=0 (not clustered)
    g1[1] = (int)(64u << 16);                       // tensor_dim0 = 64
    g1[2] = (int)((rows_avail & 0xFFFFu) << 16);    // tensor_dim1[15:0]
    g1[3] = (int)(((rows_avail >> 16) & 0xFFFFu)    // tensor_dim1[31:16]
         | (64u << 16));                            // tile_dim0 = 64
    g1[4] = 64;                                     // tile_dim1 = 64 rows (tile_dim2=0)
    g1[5] = 64;                                     // tensor_dim0_stride = 64
    g1[6] = 0;
    g1[7] = 0;
    const v4i z4 = {0, 0, 0, 0};                    // groups 2/3 unused (2D tile)
    const v8i z8 = {0, 0, 0, 0, 0, 0, 0, 0};        // trailing group: disabled
    __builtin_amdgcn_tensor_load_to_lds(g0, g1, z4, z4, z8, 0);
}

// =====================================================================
// Fused GEMM kernel:  Out = [lrelu]( (Hin [+Agg]) @ W1 + B1 ) [@ W2 + B2]
//   dbl=0 : Out = Z @ W1 + B1                  (pre-linear)
//   dbl=1 : Out = lrelu(Z @ W1 + B1) @ W2 + B2 (GIN conv MLP / post MLP)
// Block: 256 threads (8 wave32s), 64 rows per block.
// Weights (and Z when no Agg) staged via TENSOR_LOAD_TO_LDS; matmuls lower
// to v_wmma_f32_16x16x4_f32 (fp32 WMMA, exact numerics).
// =====================================================================
__global__ __launch_bounds__(256) void gin_mlp(
    const float* __restrict__ Hin, const float* __restrict__ Agg,
    const float* __restrict__ W1,  const float* __restrict__ B1,
    const float* __restrict__ W2,  const float* __restrict__ B2,
    float* __restrict__ Out, int n, int dbl)
{
    __shared__ float Zs[64 * ZPAD];   // activation tile (reused for intermediate T)
    __shared__ float W1s[64 * 64];
    __shared__ float W2s[64 * 64];
    __shared__ float B1s[64], B2s[64];

    const int tid  = threadIdx.x;
    const int row0 = blockIdx.x * 64;
    const int wv   = tid >> 5;      // wave id 0..7
    const bool zDMA = (Agg == nullptr);

    // ---- TDM staging: one wave each issues a tensor DMA (EXEC ignored) ----
    if (wv == 0) {
        tdm_load_tile(W1s, W1, 64u, false);
    } else if (wv == 1 && dbl) {
        tdm_load_tile(W2s, W2, 64u, false);
    } else if (wv == 2 && zDMA) {
        tdm_load_tile(Zs, Hin + (size_t)row0 * HD, (unsigned)(n - row0), true);
    }

    // ---- biases (tiny, manual) ----
    if (tid < 64) {
        B1s[tid] = B1[tid];
        if (dbl) B2s[tid] = B2[tid];
    }

    // ---- conv layers: stage Z = Hin + Agg manually (TDM can't fuse the add) ----
    if (!zDMA) {
        const int r  = tid >> 2;            // 0..63
        const int c0 = (tid & 3) * 16;      // 0,16,32,48
        const int row = row0 + r;
        float* zrow = &Zs[r * ZPAD];
        if (row < n) {
            const float* hp = Hin + (size_t)row * HD + c0;
            const float* ap = Agg + (size_t)row * HD + c0;
            #pragma unroll
            for (int j = 0; j < 16; ++j) zrow[c0 + j] = hp[j] + ap[j];
        } else {
            #pragma unroll
            for (int j = 0; j < 16; ++j) zrow[c0 + j] = 0.0f;
        }
    }

    // each wave waits for its own TDM (no-op where none was issued), then barrier
    __builtin_amdgcn_s_wait_tensorcnt(0);
    __syncthreads();

    const int lane = tid & 31;
    const int half = lane >> 4;     // 0 | 1
    const int l15  = lane & 15;

    // ---------------- GEMM1: T = Z @ W1 + B1 (lrelu if dbl) ----------------
    v8f acc[2];
    #pragma unroll
    for (int ti = 0; ti < 2; ++ti) {
        const int t  = wv * 2 + ti;
        const int rt = t >> 2, ct = t & 3;
        v8f c = {0.f,0.f,0.f,0.f,0.f,0.f,0.f,0.f};
        // A frag: a[v] = Z[rt*16 + (lane&15)][k + 2*half + v]
        // B frag: b[v] = W[k + 2*half + v][ct*16 + (lane&15)]
        const float* za = &Zs[(rt * 16 + l15) * ZPAD + 2 * half];
        const float* wb = &W1s[(2 * half) * 64 + ct * 16 + l15];
        #pragma unroll
        for (int k = 0; k < 64; k += 4) {
            v2f a; a.x = za[k];      a.y = za[k + 1];
            v2f b; b.x = wb[k * 64]; b.y = wb[(k + 1) * 64];
            c = __builtin_amdgcn_wmma_f32_16x16x4_f32(
                    false, a, false, b, (short)0, c, false, false);
        }
        const float bias = B1s[ct * 16 + l15];
        #pragma unroll
        for (int v = 0; v < 8; ++v) {
            float x = c[v] + bias;
            if (dbl) x = (x > 0.0f) ? x : x * SLOPE;   // LeakyReLU between the two GEMMs
            c[v] = x;
        }
        acc[ti] = c;
    }

    if (!dbl) {   // single-GEMM mode (pre-linear): store and exit
        #pragma unroll
        for (int ti = 0; ti < 2; ++ti) {
            const int t  = wv * 2 + ti;
            const int rt = t >> 2, ct = t & 3;
            #pragma unroll
            for (int v = 0; v < 8; ++v) {
                const int row = row0 + rt * 16 + v + 8 * half;
                if (row < n) Out[(size_t)row * HD + ct * 16 + l15] = acc[ti][v];
            }
        }
        return;
    }

    // ---- stash T back into Zs (all Z reads are done; barrier-separated) ----
    __syncthreads();
    #pragma unroll
    for (int ti = 0; ti < 2; ++ti) {
        const int t  = wv * 2 + ti;
        const int rt = t >> 2, ct = t & 3;
        #pragma unroll
        for (int v = 0; v < 8; ++v)
            Zs[(rt * 16 + v + 8 * half) * ZPAD + ct * 16 + l15] = acc[ti][v];
    }
    __syncthreads();

    // ---------------- GEMM2: Out = T @ W2 + B2 ----------------
    #pragma unroll
    for (int ti = 0; ti < 2; ++ti) {
        const int t  = wv * 2 + ti;
        const int rt = t >> 2, ct = t & 3;
        v8f c = {0.f,0.f,0.f,0.f,0.f,0.f,0.f,0.f};
        const float* za = &Zs[(rt * 16 + l15) * ZPAD + 2 * half];
        const float* wb = &W2s[(2 * half) * 64 + ct * 16 + l15];
        #pragma unroll
        for (int k = 0; k < 64; k += 4) {
            v2f a; a.x = za[k];      a.y = za[k + 1];
            v2f b; b.x = wb[k * 64]; b.y = wb[(k + 1) * 64];
            c = __builtin_amdgcn_wmma_f32_16x16x4_f32(
                    false, a, false, b, (short)0, c, false, false);
        }
        const float bias = B2s[ct * 16 + l15];
        #pragma unroll
        for (int v = 0; v < 8; ++v) {
            const int row = row0 + rt * 16 + v + 8 * half;
            if (row < n) Out[(size_t)row * HD + ct * 16 + l15] = c[v] + bias;
        }
    }
}

// =====================================================================
// Edge scatter: agg[dst] += h[src].  16 lanes = one edge = 256 contiguous
// bytes gathered, then 4 float atomics each (resolve in L2: agg is 25.6MB).
// =====================================================================
__global__ __launch_bounds__(256) void k_scatter(
    const float* __restrict__ h, const long long* __restrict__ src,
    const long long* __restrict__ dst, float* __restrict__ agg)
{
    const unsigned long long t =
        (unsigned long long)blockIdx.x * 256ull + threadIdx.x;
    if (t >= (unsigned long long)E_EDGES * 16ull) return;
    const unsigned e  = (unsigned)(t >> 4);
    const int      f4 = (int)(t & 15) * 4;
    const int s = (int)src[e];
    const int d = (int)dst[e];
    const float4 v = *(const float4*)(h + (size_t)s * HD + f4);
    float* p = agg + (size_t)d * HD + f4;
    atomicAdd(p + 0, v.x);
    atomicAdd(p + 1, v.y);
    atomicAdd(p + 2, v.z);
    atomicAdd(p + 3, v.w);
}

// ---------------- batchnorm: column sums / sumsq ----------------
__global__ __launch_bounds__(256) void k_colstats(
    const float* __restrict__ h, float* __restrict__ stats, int n)
{
    const int col = threadIdx.x & 63;
    const int grp = threadIdx.x >> 6;        // 0..3
    float s = 0.f, q = 0.f;
    for (int r = blockIdx.x * 4 + grp; r < n; r += gridDim.x * 4) {
        const float v = h[(size_t)r * HD + col];
        s += v; q += v * v;
    }
    __shared__ float sh[2][4][64];
    sh[0][grp][col] = s; sh[1][grp][col] = q;
    __syncthreads();
    if (grp == 0) {
        s = sh[0][0][col] + sh[0][1][col] + sh[0][2][col] + sh[0][3][col];
        q = sh[1][0][col] + sh[1][1][col] + sh[1][2][col] + sh[1][3][col];
        atomicAdd(&stats[col], s);
        atomicAdd(&stats[64 + col], q);
    }
}

__global__ void k_bn_fin(float* __restrict__ stats, const float* __restrict__ g,
                         const float* __restrict__ b, int n)
{
    const int c = threadIdx.x;
    if (c < 64) {
        const float mean  = stats[c] / (float)n;
        const float var   = stats[64 + c] / (float)n - mean * mean;  // population var (jnp.var)
        const float scale = g[c] * rsqrtf(var + BN_EPS);
        stats[128 + c] = scale;
        stats[192 + c] = b[c] - mean * scale;
    }
}

__global__ __launch_bounds__(256) void k_bn_apply(
    float* __restrict__ h, const float* __restrict__ stats, int n)
{
    const size_t i = (size_t)blockIdx.x * 256 + threadIdx.x;   // one float4 each
    if (i >= (size_t)n * 16) return;
    const int c0 = (int)(i & 15) * 4;
    const float* sc = stats + 128;
    const float* sf = stats + 192;
    float4 v = ((float4*)h)[i];
    v.x = v.x * sc[c0 + 0] + sf[c0 + 0];
    v.y = v.y * sc[c0 + 1] + sf[c0 + 1];
    v.z = v.z * sc[c0 + 2] + sf[c0 + 2];
    v.w = v.w * sc[c0 + 3] + sf[c0 + 3];
    ((float4*)h)[i] = v;
}

// =====================================================================
extern "C" void kernel_launch(void* const* d_in, const int* in_sizes, int n_in,
                              void* d_out, int out_size, void* d_ws, size_t ws_size,
                              hipStream_t stream)
{
    const float*     x       = (const float*)d_in[0];
    const long long* ei      = (const long long*)d_in[1];   // (2, E) int64
    const float*     pre_w   = (const float*)d_in[2];
    const float*     pre_b   = (const float*)d_in[3];
    const float*     conv_w1 = (const float*)d_in[4];
    const float*     conv_b1 = (const float*)d_in[5];
    const float*     conv_w2 = (const float*)d_in[6];
    const float*     conv_b2 = (const float*)d_in[7];
    const float*     bn_g    = (const float*)d_in[8];
    const float*     bn_b    = (const float*)d_in[9];
    const float*     post_w1 = (const float*)d_in[10];
    const float*     post_b1 = (const float*)d_in[11];
    const float*     post_w2 = (const float*)d_in[12];
    const float*     post_b2 = (const float*)d_in[13];

    const size_t NH = (size_t)N_NODES * HD;
    float* h     = (float*)d_ws;       // 25.6 MB
    float* agg   = h + NH;             // 25.6 MB
    float* stats = agg + NH;           // 256 floats

    const long long* src = ei;
    const long long* dst = ei + E_EDGES;

    const dim3 blk(256);
    const int  gemm_blocks    = (N_NODES + 63) / 64;
    const int  scatter_blocks = (int)(((unsigned long long)E_EDGES * 16ull + 255ull) / 256ull);
    const int  bn_blocks      = (int)(((size_t)N_NODES * 16 + 255) / 256);

    // pre-linear: h = x @ pre_w + pre_b
    gin_mlp<<<gemm_blocks, blk, 0, stream>>>(x, nullptr, pre_w, pre_b,
                                             nullptr, nullptr, h, N_NODES, 0);

    for (int l = 0; l < 3; ++l) {
        (void)hipMemsetAsync(agg, 0, NH * sizeof(float), stream);
        k_scatter<<<scatter_blocks, blk, 0, stream>>>(h, src, dst, agg);

        gin_mlp<<<gemm_blocks, blk, 0, stream>>>(h, agg,
                                                 conv_w1 + (size_t)l * HD * HD,
                                                 conv_b1 + (size_t)l * HD,
                                                 conv_w2 + (size_t)l * HD * HD,
                                                 conv_b2 + (size_t)l * HD,
                                                 h /*in-place safe*/, N_NODES, 1);
        if (l < 2) {
            (void)hipMemsetAsync(stats, 0, 256 * sizeof(float), stream);
            k_colstats<<<1024, blk, 0, stream>>>(h, stats, N_NODES);
            k_bn_fin<<<1, 64, 0, stream>>>(stats, bn_g + (size_t)l * HD,
                                           bn_b + (size_t)l * HD, N_NODES);
            k_bn_apply<<<bn_blocks, blk, 0, stream>>>(h, stats, N_NODES);
        }
    }

    // post MLP: out = lrelu(h @ pw1 + pb1) @ pw2 + pb2  (same fused structure)
    gin_mlp<<<gemm_blocks, blk, 0, stream>>>(h, nullptr, post_w1, post_b1,
                                             post_w2, post_b2,
                                             (float*)d_out, N_NODES, 1);
}